// mLSTMCell_57286273794823
// MI455X (gfx1250) — compile-verified
//
#include <hip/hip_runtime.h>

typedef __attribute__((ext_vector_type(2))) float v2f;
typedef __attribute__((ext_vector_type(4))) float v4f;
typedef __attribute__((ext_vector_type(8))) float v8f;

#define HDIM 1024
#define BDIM 64

// ---------------------------------------------------------------------------
// Kernel 1: all projections via V_WMMA_F32_16X16X4_F32.
// Logical output columns: [0,3072) gates (i|f|o), [3072,4096) q,
// [4096,5120) k, [5120,6144) v.  One wave per 16x16 tile.
// Tiles: 4 M-tiles x 384 N-tiles = 1536 waves = 192 blocks x 8 waves.
// Weights (~36 MB) stay resident in the 192 MB L2 because the big c-stream
// in kernel 2 is marked non-temporal.
// ---------------------------------------------------------------------------
__global__ __launch_bounds__(256) void proj_gemm_wmma(
    const float* __restrict__ input, const float* __restrict__ hprev,
    const float* __restrict__ w_ih, const float* __restrict__ w_hh,
    const float* __restrict__ bias,
    const float* __restrict__ wq_w, const float* __restrict__ wq_b,
    const float* __restrict__ wk_w, const float* __restrict__ wk_b,
    const float* __restrict__ wv_w, const float* __restrict__ wv_b,
    float* __restrict__ ig, float* __restrict__ fg, float* __restrict__ og,
    float* __restrict__ qo, float* __restrict__ ko, float* __restrict__ vo)
{
    const int K = HDIM;
    const int wid  = blockIdx.x * (blockDim.x >> 5) + (threadIdx.x >> 5);
    const int lane = threadIdx.x & 31;
    const int mt = wid & 3;         // 4 M tiles (64 rows)
    const int nt = wid >> 2;        // 0..383 N tiles
    const int m0 = mt << 4;
    const int mr = lane & 15;       // A row within tile / B column within tile
    const int kb = (lane >> 4) << 1;// K sub-offset: 0 (lanes 0-15) or 2 (16-31)

    const float* X1; const float* W1;
    const float* X2 = nullptr; const float* W2 = nullptr;
    const float* bptr; float* dst; int col; int act;

    if (nt < 192) {                       // gates region, cols [0,3072)
        const int n0 = nt << 4;
        X1 = input; W1 = w_ih + (size_t)n0 * K;
        X2 = hprev; W2 = w_hh + (size_t)n0 * K;
        bptr = bias + n0;
        const int region = n0 >> 10;      // 0=i, 1=f, 2=o
        col  = n0 & 1023;
        dst  = (region == 0) ? ig : (region == 1) ? fg : og;
        act  = (region == 2) ? 2 : 1;     // sigmoid for o, exp for i,f
    } else {                              // q/k/v regions
        const int r  = (nt - 192) >> 6;   // 0=q, 1=k, 2=v
        const int n0 = ((nt - 192) & 63) << 4;
        X1 = input;
        W1 = ((r == 0) ? wq_w : (r == 1) ? wk_w : wv_w) + (size_t)n0 * K;
        bptr = ((r == 0) ? wq_b : (r == 1) ? wk_b : wv_b) + n0;
        col = n0;
        dst = (r == 0) ? qo : (r == 1) ? ko : vo;
        act = 0;
    }

    v8f acc = {};
    {
        const float* xr = X1 + (size_t)(m0 + mr) * K;   // A: row m0+mr
        const float* wr = W1 + (size_t)mr * K;          // B: col (lane&15)
        for (int kk = 0; kk < K; kk += 4) {
            v2f a, b;
            a.x = xr[kk + kb]; a.y = xr[kk + kb + 1];
            b.x = wr[kk + kb]; b.y = wr[kk + kb + 1];
            acc = __builtin_amdgcn_wmma_f32_16x16x4_f32(
                false, a, false, b, (short)0, acc, false, false);
        }
    }
    if (X2) {                            // second accumulation pass for gates
        const float* xr = X2 + (size_t)(m0 + mr) * K;
        const float* wr = W2 + (size_t)mr * K;
        for (int kk = 0; kk < K; kk += 4) {
            v2f a, b;
            a.x = xr[kk + kb]; a.y = xr[kk + kb + 1];
            b.x = wr[kk + kb]; b.y = wr[kk + kb + 1];
            acc = __builtin_amdgcn_wmma_f32_16x16x4_f32(
                false, a, false, b, (short)0, acc, false, false);
        }
    }

    // Epilogue: C/D layout -> VGPR r holds row m0+r (lanes 0-15) / m0+8+r (16-31)
    const int n     = lane & 15;
    const int mbase = m0 + ((lane >> 4) << 3);
    const float bv  = bptr[n];
    #pragma unroll
    for (int r = 0; r < 8; ++r) {
        float y = acc[r] + bv;
        if (act == 1)      y = __expf(y);
        else if (act == 2) y = 1.0f / (1.0f + __expf(-y));
        dst[(size_t)(mbase + r) * HDIM + col + n] = y;
    }
}

// ---------------------------------------------------------------------------
// Kernel 2: fused streaming pass over c (the bandwidth-bound part:
// 512 MB @ 23.3 TB/s ~= 22 us floor).
// grid = (8 h-slices, 64 batches), block = 256 threads, one v4f per thread
// covers the full k-range (256 * 4 = 1024).  Exactly one read of c and one
// write of c_new, both NON-TEMPORAL (the 512 MB stream must not evict the
// L2-resident weights/gates); q-weighted partial sums live in registers.
// ---------------------------------------------------------------------------
__global__ __launch_bounds__(256) void c_update_stream(
    const float* __restrict__ c_in,
    const float* __restrict__ ig, const float* __restrict__ fg,
    const float* __restrict__ qv, const float* __restrict__ kv_,
    const float* __restrict__ vv,
    float* __restrict__ c_out, float* __restrict__ hacc /* [8][64][1024] */)
{
    const int s = blockIdx.x;        // h-slice (128 rows each)
    const int b = blockIdx.y;        // batch
    const int t = threadIdx.x;       // v4f index over k

    const v4f kvec = ((const v4f*)(kv_ + (size_t)b * HDIM))[t];
    v4f acc = {0.f, 0.f, 0.f, 0.f};

    const int h0 = s * 128;
    const v4f* crow = (const v4f*)(c_in  + ((size_t)b * HDIM + h0) * HDIM) + t;
    v4f*       orow = (v4f*)      (c_out + ((size_t)b * HDIM + h0) * HDIM) + t;
    const float* igr = ig + (size_t)b * HDIM + h0;
    const float* fgr = fg + (size_t)b * HDIM + h0;
    const float* qr  = qv + (size_t)b * HDIM + h0;
    const float* vr  = vv + (size_t)b * HDIM + h0;

    #pragma unroll 4
    for (int hh = 0; hh < 128; ++hh) {
        const float f   = fgr[hh];            // uniform -> scalar loads
        const float ivg = igr[hh] * vr[hh];
        const float qg  = qr[hh];

        v4f cv = __builtin_nontemporal_load(crow);   // TH=NT: read-once stream
        v4f cn;
        cn.x = f * cv.x + ivg * kvec.x;
        cn.y = f * cv.y + ivg * kvec.y;
        cn.z = f * cv.z + ivg * kvec.z;
        cn.w = f * cv.w + ivg * kvec.w;
        __builtin_nontemporal_store(cn, orow);       // TH=NT: write-once stream

        acc.x += qg * cn.x; acc.y += qg * cn.y;
        acc.z += qg * cn.z; acc.w += qg * cn.w;

        crow += HDIM / 4; orow += HDIM / 4;
    }
    ((v4f*)(hacc + ((size_t)s * BDIM + b) * HDIM))[t] = acc;
}

// ---------------------------------------------------------------------------
// Kernel 3: reduce the 8 slice partials, apply the (pre-sigmoided) o gate.
// ---------------------------------------------------------------------------
__global__ __launch_bounds__(256) void h_finish(
    const float* __restrict__ hacc, const float* __restrict__ og,
    float* __restrict__ h_new)
{
    const int idx = blockIdx.x * blockDim.x + threadIdx.x;  // 0..65535
    float s = 0.f;
    #pragma unroll
    for (int sl = 0; sl < 8; ++sl) s += hacc[(size_t)sl * (BDIM * HDIM) + idx];
    h_new[idx] = og[idx] * s;
}

// ---------------------------------------------------------------------------
extern "C" void kernel_launch(void* const* d_in, const int* in_sizes, int n_in,
                              void* d_out, int out_size, void* d_ws, size_t ws_size,
                              hipStream_t stream)
{
    (void)in_sizes; (void)n_in; (void)out_size; (void)ws_size;

    const float* input = (const float*)d_in[0];
    const float* hprev = (const float*)d_in[1];
    const float* c     = (const float*)d_in[2];
    const float* w_ih  = (const float*)d_in[3];
    const float* w_hh  = (const float*)d_in[4];
    const float* bias  = (const float*)d_in[5];
    const float* wq_w  = (const float*)d_in[6];
    const float* wq_b  = (const float*)d_in[7];
    const float* wk_w  = (const float*)d_in[8];
    const float* wk_b  = (const float*)d_in[9];
    const float* wv_w  = (const float*)d_in[10];
    const float* wv_b  = (const float*)d_in[11];

    float* out   = (float*)d_out;
    float* h_new = out;                       // [64, 1024]
    float* c_new = out + (size_t)BDIM * HDIM; // [64, 1024, 1024]

    float* w    = (float*)d_ws;               // all f32 scratch
    float* ig   = w;                          // 65536 each
    float* fg   = w + 1 * 65536;
    float* og   = w + 2 * 65536;
    float* qo   = w + 3 * 65536;
    float* ko   = w + 4 * 65536;
    float* vo   = w + 5 * 65536;
    float* hacc = w + 6 * 65536;              // 8 * 65536 floats

    proj_gemm_wmma<<<192, 256, 0, stream>>>(
        input, hprev, w_ih, w_hh, bias,
        wq_w, wq_b, wk_w, wk_b, wv_w, wv_b,
        ig, fg, og, qo, ko, vo);

    c_update_stream<<<dim3(8, 64), 256, 0, stream>>>(
        c, ig, fg, qo, ko, vo, c_new, hacc);

    h_finish<<<256, 256, 0, stream>>>(hacc, og, h_new);
}